// SoftRED_Loss_17514876633824
// MI455X (gfx1250) — compile-verified
//
#include <hip/hip_runtime.h>
#include <hip/hip_bf16.h>

// ---------------------------------------------------------------------------
// SoftRED loss:  matching = -(A @ B^T)  (512x512, K=1024, f32)  then a soft
// restricted-edit-distance DP over the (513)x(513) grid, returning D[M][N]/M.
//
// Kernel 1: f32 WMMA GEMM (v_wmma_f32_16x16x4_f32) computing Cd = +A@B^T in
//           DIAGONAL-MAJOR layout (the DP negates at use: add -> sub, free).
//           Inner loop is pure global_load_b64 + v_wmma (no sign-flip VALU).
// Kernel 2: single-workgroup anti-diagonal DP, 512 threads (16 wave32),
//           5 rolling LDS diagonal buffers, 1 barrier per diagonal,
//           hardware exp/log for the soft-min, prefetched coalesced C loads.
// ---------------------------------------------------------------------------

#define MM 512
#define NN 512
#define KK 1024
#define BG 2.0f
#define SWAPC 2.0f
#define GAMMA 0.01f
#define INVG 100.0f

typedef __attribute__((ext_vector_type(2))) float v2f;
typedef __attribute__((ext_vector_type(8))) float v8f;

// Offset of anti-diagonal k (k = row+col, 0..1022) in diagonal-major storage.
// Diagonal k holds rows r in [max(0,k-511), min(k,511)], stored at off(k)+r.
__device__ __forceinline__ int diag_off(int k) {
    return (k <= 512) ? (k * (k + 1)) / 2
                      : 262144 - ((1023 - k) * (1024 - k)) / 2;
}

// ---------------------------------------------------------------------------
// GEMM: each wave32 computes one 16x16 tile of Cd = +A @ B^T using
// V_WMMA_F32_16X16X4_F32 over K in steps of 4 (256 WMMA per tile).
// A-frag (16x4 f32): lane l -> M = l%16 ; VGPR0/1 hold K = {0,1} (lanes 0-15)
// or K = {2,3} (lanes 16-31).  B-frag (4x16) mirrors it with N = l%16.
// ---------------------------------------------------------------------------
__global__ __launch_bounds__(256) void softred_gemm_wmma(
    const float* __restrict__ A,   // [512][1024]
    const float* __restrict__ B,   // [512][1024]
    float* __restrict__ Cd)        // [262144] diagonal-major, +A@B^T
{
    const int lane  = threadIdx.x & 31;
    const int wave  = threadIdx.x >> 5;
    const int tile  = blockIdx.x * 8 + wave;    // 0..1023 (32x32 tiles)
    const int tm    = (tile >> 5) << 4;         // tile row base
    const int tn    = (tile & 31) << 4;         // tile col base
    const int m16   = lane & 15;
    const int half  = lane >> 4;                // 0: K={0,1}, 1: K={2,3}

    const float2* __restrict__ Ar =
        (const float2*)(A + (size_t)(tm + m16) * KK);
    const float2* __restrict__ Br =
        (const float2*)(B + (size_t)(tn + m16) * KK);

    v8f acc = {};
    // k2 = k/2; K advances by 4 per iteration -> k2 += 2
    #pragma unroll 4
    for (int k2 = 0; k2 < KK / 2; k2 += 2) {
        float2 af = Ar[k2 + half];
        float2 bf = Br[k2 + half];
        v2f a; a.x = af.x; a.y = af.y;
        v2f b; b.x = bf.x; b.y = bf.y;
        acc = __builtin_amdgcn_wmma_f32_16x16x4_f32(
            false, a, false, b, (short)0, acc, false, false);
    }

    // C/D layout: VGPR r, lanes 0-15 -> M = r, lanes 16-31 -> M = r+8; N = l%16
    const int Cc = tn + m16;
    #pragma unroll
    for (int r = 0; r < 8; ++r) {
        const int R = tm + r + (half << 3);
        Cd[diag_off(R + Cc) + R] = acc[r];
    }
}

// ---------------------------------------------------------------------------
// DP wavefront.  Thread t owns grid row i = t+1.  Diagonal d = i+j runs
// 2..1024.  buf[d%5][i] holds D on diagonal d at row i (index 0 = row 0
// boundary).  One __syncthreads per diagonal is sufficient: read set
// {d-1,d-2,d-4} mod 5 never intersects write set {d} mod 5, and the slot
// being overwritten (d-5) was last read one barrier ago.
// Cd holds +A@B^T, so "matching" values enter negated (adds become subs).
// ---------------------------------------------------------------------------
__global__ __launch_bounds__(512) void softred_dp(
    const float* __restrict__ Cd,  // diagonal-major +A@B^T
    float* __restrict__ out)
{
    __shared__ float buf[5][MM + 1];
    const int t = threadIdx.x;     // 0..511
    const int i = t + 1;           // grid row 1..512

    if (t == 0) { buf[0][0] = 0.0f; buf[1][0] = BG; }
    if (t == 1) { buf[1][1] = BG; }

    // c_cur = +dot at cell (i,j) for current diagonal; c_prev = +dot(i,j-1);
    // cm_cur = +dot(i-1,j).  Preload for d = 2 (only thread 0 active).
    float c_cur  = (t == 0) ? Cd[0] : 0.0f;
    float c_prev = 0.0f;
    float cm_cur = 0.0f;
    __syncthreads();

    for (int d = 2; d <= MM + NN; ++d) {
        const int  j   = d - i;
        const bool act = (j >= 1) && (j <= NN);
        const bool v4  = act && (j >= 2) && (i >= 2);

        // --- prefetch C for diagonal d+1 (coalesced: index = off + t) ---
        float c_nxt = 0.0f, cm_nxt = 0.0f;
        if (d < MM + NN) {
            const int jn = j + 1;
            if (jn >= 1 && jn <= NN)
                c_nxt  = Cd[diag_off(d - 1) + t];        // dot[i-1][jn-1]
            if (jn >= 2 && jn <= NN && i >= 2)
                cm_nxt = Cd[diag_off(d - 2) + t - 1];    // dot[i-2][jn-1]
        }

        // --- soft-min recurrence (matching = -dot) ---
        float dval = 0.0f;
        if (act) {
            const int bm1 = (d - 1) % 5;
            const int bm2 = (d - 2) % 5;
            const int bm4 = (d + 1) % 5;   // == (d-4) mod 5
            const float t1 = buf[bm2][i - 1] - c_cur;        // diag
            const float t2 = buf[bm1][i - 1] + BG;           // up
            const float t3 = buf[bm1][i]     + BG;           // left
            float m = fminf(fminf(t1, t2), t3);
            float t4 = 0.0f;
            if (v4) {
                t4 = buf[bm4][i - 2] - cm_cur - c_prev + SWAPC;
                m = fminf(m, t4);
            }
            float s = __expf((m - t1) * INVG)
                    + __expf((m - t2) * INVG)
                    + __expf((m - t3) * INVG);
            if (v4) s += __expf((m - t4) * INVG);
            dval = m - GAMMA * __logf(s);
            buf[d % 5][i] = dval;
        } else if (i == d) {
            // boundary D[d][0] = d*bg (only reachable for d <= 512)
            buf[d % 5][i] = (float)d * BG;
        }
        if (t == 0 && d <= NN)
            buf[d % 5][0] = (float)d * BG;   // boundary D[0][d]

        if (d == MM + NN && t == MM - 1)
            out[0] = dval * (1.0f / (float)MM);

        __syncthreads();
        c_prev = c_cur; c_cur = c_nxt; cm_cur = cm_nxt;
    }
}

// ---------------------------------------------------------------------------
extern "C" void kernel_launch(void* const* d_in, const int* in_sizes, int n_in,
                              void* d_out, int out_size, void* d_ws, size_t ws_size,
                              hipStream_t stream) {
    const float* A = (const float*)d_in[0];   // centers_a [512,1024] f32
    const float* B = (const float*)d_in[1];   // centers_b [512,1024] f32
    float* Cd  = (float*)d_ws;                // 512*512 f32 = 1 MB scratch
    float* out = (float*)d_out;               // scalar f32

    // 1024 tiles of 16x16, 8 waves (tiles) per 256-thread block -> 128 blocks
    softred_gemm_wmma<<<128, 256, 0, stream>>>(A, B, Cd);
    // single 512-thread workgroup runs the 1023-step wavefront
    softred_dp<<<1, 512, 0, stream>>>(Cd, out);
}